// SpecLayer_15530601742865
// MI455X (gfx1250) — compile-verified
//
#include <hip/hip_runtime.h>
#include <hip/hip_bf16.h>

// Problem constants (match reference)
#define G_  128
#define N_  2000
#define E_  64
#define I_  128
#define K_  128
#define GN_ (G_ * N_)
#define BN_EPS 1e-5f
#define ROWS_ 80              // eigen rows staged per TDM tile (2000 = 25 * 80)
#define STAGES_ 25

typedef __attribute__((ext_vector_type(2))) float v2f;
typedef __attribute__((ext_vector_type(8))) float v8f;
typedef __attribute__((ext_vector_type(4))) unsigned int u32x4;
typedef __attribute__((ext_vector_type(4))) int i32x4;
typedef __attribute__((ext_vector_type(8))) int i32x8;

// ---------------------------------------------------------------------------
// TDM: issue a 2D tile load (rows x cols f32, row-major, fully in-bounds)
// from global memory into LDS. Descriptor per CDNA5 ISA.
// Must be executed by exactly one wave; completion via s_wait_tensorcnt.
// ---------------------------------------------------------------------------
__device__ __forceinline__ void tdm_load_2d(const void* gaddr, void* ldsdst,
                                            unsigned rows, unsigned cols,
                                            unsigned row_stride_elems) {
    const unsigned long long ga = (unsigned long long)(uintptr_t)gaddr;
    const unsigned ldsoff = (unsigned)(uintptr_t)ldsdst;   // low 32 bits = LDS offset

    u32x4 g0;
    g0.x = 1u;                                   // count=1, user descriptor
    g0.y = ldsoff;                               // lds_addr
    g0.z = (unsigned)(ga & 0xFFFFFFFFu);         // global_addr[31:0]
    g0.w = (unsigned)((ga >> 32) & 0x01FFFFFFu)  // global_addr[56:32]
         | (2u << 30);                           // type = 2 ("image")

    i32x8 g1;
    g1[0] = (int)(2u << 16);                     // workgroup_mask=0, data_size=2 (4B)
    g1[1] = (int)((cols & 0xFFFFu) << 16);       // tensor_dim0[15:0] @ bits 63:48
    g1[2] = (int)((cols >> 16) & 0xFFFFu)        // tensor_dim0[31:16]
          | (int)((rows & 0xFFFFu) << 16);       // tensor_dim1[15:0]
    g1[3] = (int)((rows >> 16) & 0xFFFFu)        // tensor_dim1[31:16]
          | (int)((cols & 0xFFFFu) << 16);       // tile_dim0
    g1[4] = (int)(rows & 0xFFFFu);               // tile_dim1 (tile_dim2 = 0)
    g1[5] = (int)row_stride_elems;               // tensor_dim0_stride[31:0]
    g1[6] = 0;                                   // stride hi / dim1_stride lo
    g1[7] = 0;

    i32x4 g2 = {0, 0, 0, 0};
    i32x4 g3 = {0, 0, 0, 0};

#if __clang_major__ >= 23
    i32x8 g4 = {0, 0, 0, 0, 0, 0, 0, 0};
    __builtin_amdgcn_tensor_load_to_lds(g0, g1, g2, g3, g4, 0);
#else
    __builtin_amdgcn_tensor_load_to_lds(g0, g1, g2, g3, 0);
#endif
}

// ---------------------------------------------------------------------------
// Kernel 0: zero the BN accumulators (sums[128], sqs[128])
// ---------------------------------------------------------------------------
__global__ __launch_bounds__(256) void k_zero(float* __restrict__ p) {
    p[threadIdx.x] = 0.0f;   // 256 floats: sums then sqs
}

// ---------------------------------------------------------------------------
// Kernel 1: spec_in[g] = feat[g]^T (I x N) @ eigen[g] (N x E)  -> [G, I, E]
// One block per graph, 8 waves. Eigen is staged through LDS with the Tensor
// Data Mover, double-buffered (wave 0 issues, all waves consume); the DMA of
// stage s+1 overlaps the WMMA compute of stage s.
// ---------------------------------------------------------------------------
__global__ __launch_bounds__(256)
void k_project(const float* __restrict__ feat, const float* __restrict__ eigen,
               float* __restrict__ spec_in) {
    const int g    = blockIdx.x;
    const int wave = threadIdx.x >> 5;
    const int lane = threadIdx.x & 31;
    const int lm   = lane & 15;
    const int kh   = (lane >> 4) << 1;          // 0 or 2 (K half select)
    const int m0   = wave * 16;                 // I-strip base

    __shared__ float Ebuf[2][ROWS_ * E_];       // 2 x 20 KB eigen tiles

    const float* fg = feat  + (size_t)g * N_ * I_;
    const float* eg = eigen + (size_t)g * N_ * E_;

    if (wave == 0)
        tdm_load_2d(eg, &Ebuf[0][0], ROWS_, E_, E_);

    v8f acc[4] = {};                            // E tiles n0 = 0,16,32,48
    for (int s = 0; s < STAGES_; ++s) {
        if (wave == 0)
            __builtin_amdgcn_s_wait_tensorcnt(0);   // stage s tile landed
        __syncthreads();
        if (wave == 0 && s + 1 < STAGES_)
            tdm_load_2d(eg + (size_t)(s + 1) * ROWS_ * E_,
                        &Ebuf[(s + 1) & 1][0], ROWS_, E_, E_);

        const float* Eb   = &Ebuf[s & 1][0];
        const float* frow = fg + (size_t)s * ROWS_ * I_;

        // prefetch next feature chunk (40 KB) while we compute this one
        if (s + 1 < STAGES_)
            __builtin_prefetch(frow + (size_t)ROWS_ * I_ + threadIdx.x * 40, 0, 0);

        for (int kr = 0; kr < ROWS_; kr += 4) {
            const int ka = kr + kh;
            v2f a;
            a.x = frow[(size_t)ka * I_ + m0 + lm];
            a.y = frow[(size_t)(ka + 1) * I_ + m0 + lm];
#pragma unroll
            for (int t = 0; t < 4; ++t) {
                v2f b;
                b.x = Eb[ka * E_ + t * 16 + lm];
                b.y = Eb[(ka + 1) * E_ + t * 16 + lm];
                acc[t] = __builtin_amdgcn_wmma_f32_16x16x4_f32(
                    false, a, false, b, (short)0, acc[t], false, false);
            }
        }
        __syncthreads();    // all waves done with Ebuf[s&1] before stage s+2 reuses it
    }

    // D layout: lanes 0-15 -> M=r, N=lane ; lanes 16-31 -> M=r+8, N=lane-16
    float* sg = spec_in + (size_t)g * I_ * E_;
    const int mrow = m0 + (lane >> 4) * 8;
#pragma unroll
    for (int r = 0; r < 8; ++r)
#pragma unroll
        for (int t = 0; t < 4; ++t)
            sg[(size_t)(mrow + r) * E_ + t * 16 + lm] = acc[t][r];
}

// ---------------------------------------------------------------------------
// Kernel 2: per-channel MLP over all graphs.
//   H = relu(S_i @ W1_i + b1_i)   S_i: [G=128, E=64], W1_i: [64,128]
//   O = H @ W2_i + b2_i           W2_i: [128,64]  -> spec_out[:, i, :]
// One block per channel i. LDS round-trip converts the GEMM1 D-layout into
// the GEMM2 A-layout.
// ---------------------------------------------------------------------------
__global__ __launch_bounds__(256)
void k_mlp(const float* __restrict__ spec_in, const float* __restrict__ w1,
           const float* __restrict__ b1, const float* __restrict__ w2,
           const float* __restrict__ b2, float* __restrict__ spec_out) {
    const int i    = blockIdx.x;
    const int wave = threadIdx.x >> 5;
    const int lane = threadIdx.x & 31;
    const int lm   = lane & 15;
    const int kh   = (lane >> 4) << 1;
    const int m0   = wave * 16;                 // G-strip base

    __shared__ float Hs[K_ * K_];               // 128x128 f32 = 64 KB

    const float* W1 = w1 + (size_t)i * E_ * K_; // [64 x 128]
    const float* W2 = w2 + (size_t)i * K_ * E_; // [128 x 64]

    // --- GEMM1: M=128(G), K=64(E), N=128 ---
    v8f acc[8] = {};
    for (int k0 = 0; k0 < E_; k0 += 4) {
        const int ka = k0 + kh;
        const v2f a = *(const v2f*)(spec_in + (size_t)(m0 + lm) * (I_ * E_)
                                    + (size_t)i * E_ + ka);   // b64 load
#pragma unroll
        for (int t = 0; t < 8; ++t) {
            v2f b;
            b.x = W1[(size_t)ka * K_ + t * 16 + lm];
            b.y = W1[(size_t)(ka + 1) * K_ + t * 16 + lm];
            acc[t] = __builtin_amdgcn_wmma_f32_16x16x4_f32(
                false, a, false, b, (short)0, acc[t], false, false);
        }
    }

    // bias + relu -> LDS (layout conversion for GEMM2 A operand)
    const int mrow = m0 + (lane >> 4) * 8;
#pragma unroll
    for (int t = 0; t < 8; ++t) {
        const float bias = b1[(size_t)i * K_ + t * 16 + lm];
#pragma unroll
        for (int r = 0; r < 8; ++r) {
            float hv = acc[t][r] + bias;
            Hs[(mrow + r) * K_ + t * 16 + lm] = hv > 0.0f ? hv : 0.0f;
        }
    }
    __syncthreads();

    // --- GEMM2: M=128(G), K=128, N=64(E) ---
    v8f oc[4] = {};
    for (int k0 = 0; k0 < K_; k0 += 4) {
        const int ka = k0 + kh;
        const v2f a = *(const v2f*)&Hs[(m0 + lm) * K_ + ka];  // ds_load_b64
#pragma unroll
        for (int t = 0; t < 4; ++t) {
            v2f b;
            b.x = W2[(size_t)ka * E_ + t * 16 + lm];
            b.y = W2[(size_t)(ka + 1) * E_ + t * 16 + lm];
            oc[t] = __builtin_amdgcn_wmma_f32_16x16x4_f32(
                false, a, false, b, (short)0, oc[t], false, false);
        }
    }
#pragma unroll
    for (int t = 0; t < 4; ++t) {
        const float bias = b2[(size_t)i * E_ + t * 16 + lm];
#pragma unroll
        for (int r = 0; r < 8; ++r)
            spec_out[(size_t)(mrow + r) * (I_ * E_) + (size_t)i * E_ + t * 16 + lm] =
                oc[t][r] + bias;
    }
}

// ---------------------------------------------------------------------------
// Kernel 3: h[g] = eigen[g] (N x E) @ spec_out[g]^T (E x I) -> d_out scratch,
// plus per-channel partial BN sums via LDS reduce + one atomicAdd per channel
// per block. spec_out[g] (32 KB) is staged into LDS by the Tensor Data Mover.
// Grid: (16 row-chunks of 128, G). Block: 256 threads (8 waves x 16 rows).
// ---------------------------------------------------------------------------
__global__ __launch_bounds__(256)
void k_inverse(const float* __restrict__ eigen, const float* __restrict__ spec_out,
               float* __restrict__ h, float* __restrict__ sums,
               float* __restrict__ sqs) {
    const int g     = blockIdx.y;
    const int chunk = blockIdx.x;
    const int wave  = threadIdx.x >> 5;
    const int lane  = threadIdx.x & 31;
    const int lm    = lane & 15;
    const int kh    = (lane >> 4) << 1;

    __shared__ float Bs[I_ * E_];   // spec_out[g] as [I][E], 32 KB
    __shared__ float redS[8 * I_];  // per-wave channel partials
    __shared__ float redQ[8 * I_];

    // Stage spec_out[g] into LDS via TDM (one shot, whole 128x64 tile)
    const float* sg = spec_out + (size_t)g * I_ * E_;
    if (wave == 0) {
        tdm_load_2d(sg, &Bs[0], I_, E_, E_);
        __builtin_amdgcn_s_wait_tensorcnt(0);
    }
    __syncthreads();

    const int row0 = chunk * 128 + wave * 16;
    const float* eg = eigen + (size_t)g * N_ * E_;

    const int  arow   = row0 + lm;
    const bool avalid = arow < N_;

    v8f acc[8] = {};                // I tiles n0 = 0..112
    for (int k0 = 0; k0 < E_; k0 += 4) {
        const int ka = k0 + kh;
        v2f a = {0.0f, 0.0f};
        if (avalid)
            a = *(const v2f*)(eg + (size_t)arow * E_ + ka);   // b64 load
#pragma unroll
        for (int t = 0; t < 8; ++t) {
            // B[k, n] = spec_out[g, n, k] = Bs[n*E + k]; contiguous pair
            const v2f b = *(const v2f*)&Bs[(t * 16 + lm) * E_ + ka];
            acc[t] = __builtin_amdgcn_wmma_f32_16x16x4_f32(
                false, a, false, b, (short)0, acc[t], false, false);
        }
    }

    // Store h + accumulate per-channel partial sums.
    // Rows >= N_ produced zeros (A was zeroed) so they add nothing.
    const int mbase = row0 + (lane >> 4) * 8;
#pragma unroll
    for (int t = 0; t < 8; ++t) {
        float s = 0.0f, q = 0.0f;
#pragma unroll
        for (int r = 0; r < 8; ++r) {
            const float v  = acc[t][r];
            const int  row = mbase + r;
            if (row < N_)
                h[((size_t)g * N_ + row) * I_ + t * 16 + lm] = v;
            s += v;
            q += v * v;
        }
        s += __shfl_xor(s, 16);
        q += __shfl_xor(q, 16);
        if (lane < 16) {
            redS[wave * I_ + t * 16 + lm] = s;
            redQ[wave * I_ + t * 16 + lm] = q;
        }
    }
    __syncthreads();

    if (threadIdx.x < I_) {
        float s = 0.0f, q = 0.0f;
#pragma unroll
        for (int w = 0; w < 8; ++w) {
            s += redS[w * I_ + threadIdx.x];
            q += redQ[w * I_ + threadIdx.x];
        }
        atomicAdd(&sums[threadIdx.x], s);
        atomicAdd(&sqs[threadIdx.x],  q);
    }
}

// ---------------------------------------------------------------------------
// Kernel 4: BatchNorm(train stats) + ReLU + residual, in place on d_out.
//   out = feat + relu((h - mean) * rsqrt(var + eps) * gamma + beta)
// ---------------------------------------------------------------------------
__global__ __launch_bounds__(256)
void k_bn(const float* __restrict__ feat, float* __restrict__ h,
          const float* __restrict__ sums, const float* __restrict__ sqs,
          const float* __restrict__ gamma, const float* __restrict__ beta) {
    const size_t idx = ((size_t)blockIdx.x * 256 + threadIdx.x) * 4;
    if (idx >= (size_t)GN_ * I_) return;
    const float invGN = 1.0f / (float)GN_;

    float4 hv = *(float4*)&h[idx];
    float4 fv = *(const float4*)&feat[idx];
    float rr[4] = {hv.x, hv.y, hv.z, hv.w};
    float ff[4] = {fv.x, fv.y, fv.z, fv.w};
    const int c0 = (int)(idx & (I_ - 1));   // I_ divides the float4 stride

#pragma unroll
    for (int j = 0; j < 4; ++j) {
        const int c    = c0 + j;
        const float mean = sums[c] * invGN;
        const float var  = sqs[c] * invGN - mean * mean;
        float x = (rr[j] - mean) * rsqrtf(var + BN_EPS) * gamma[c] + beta[c];
        x = x > 0.0f ? x : 0.0f;
        rr[j] = ff[j] + x;
    }
    float4 ov; ov.x = rr[0]; ov.y = rr[1]; ov.z = rr[2]; ov.w = rr[3];
    *(float4*)&h[idx] = ov;
}

// ---------------------------------------------------------------------------
extern "C" void kernel_launch(void* const* d_in, const int* in_sizes, int n_in,
                              void* d_out, int out_size, void* d_ws, size_t ws_size,
                              hipStream_t stream) {
    const float* feature = (const float*)d_in[0];   // [G*N, I]
    const float* eigen   = (const float*)d_in[1];   // [G, N, E]
    const float* w1      = (const float*)d_in[2];   // [I, E, K]
    const float* b1      = (const float*)d_in[3];   // [I, K]
    const float* w2      = (const float*)d_in[4];   // [I, K, E]
    const float* b2      = (const float*)d_in[5];   // [I, E]
    const float* gamma   = (const float*)d_in[6];   // [I]
    const float* beta    = (const float*)d_in[7];   // [I]
    float* out = (float*)d_out;                     // [G*N, I]; used as h scratch

    char* ws = (char*)d_ws;
    float* spec_in  = (float*)ws;                                   // 4 MB
    float* spec_out = (float*)(ws + (size_t)G_ * I_ * E_ * 4);      // 4 MB
    float* sums     = (float*)(ws + 2 * (size_t)G_ * I_ * E_ * 4);  // 128 f32
    float* sqs      = sums + I_;                                    // 128 f32

    k_zero<<<1, 256, 0, stream>>>(sums);

    k_project<<<G_, 256, 0, stream>>>(feature, eigen, spec_in);

    k_mlp<<<I_, 256, 0, stream>>>(spec_in, w1, b1, w2, b2, spec_out);

    k_inverse<<<dim3(16, G_), 256, 0, stream>>>(eigen, spec_out, out, sums, sqs);

    const size_t total = (size_t)GN_ * I_;
    k_bn<<<(int)(total / 4 / 256), 256, 0, stream>>>(feature, out, sums, sqs,
                                                     gamma, beta);
}